// WKVTorch_23716809408974
// MI455X (gfx1250) — compile-verified
//
#include <hip/hip_runtime.h>
#include <stdint.h>

// Problem dims (fixed by reference)
#define B_    8
#define S_    2048
#define E_    2048
// Tiling
#define EW    256          // channels per block (== blockDim.x)
#define TT    16           // timesteps per LDS tile
#define LC    128          // chunk length (timesteps per thread in pass 1/3)
#define NC    (S_ / LC)    // 16 chunks
#define TILES (LC / TT)    // 8 tiles per chunk
#define EB    (E_ / EW)    // 8 blocks along E

// ---------------------------------------------------------------------------
// Async-copy a TT x EW tile of k and v into LDS (double buffered).
// GVS addressing: mem = SGPR64 base + VGPR32 byte offset.
// Each thread issues 8 x b128 async ops per tile -> 8 outstanding per wave.
// ---------------------------------------------------------------------------
__device__ __forceinline__ void issue_tile(const float* kb, const float* vb,
                                           float (&sk)[2][TT][EW],
                                           float (&sv)[2][TT][EW],
                                           int buf, int tile, int tid)
{
    const uint64_t kba = (uint64_t)(uintptr_t)kb;
    const uint64_t vba = (uint64_t)(uintptr_t)vb;
#pragma unroll
    for (int i = 0; i < 4; ++i) {
        const int j   = i * EW + tid;        // 16-byte slot id within tile (0..1023)
        const int row = j >> 6;              // 64 slots of 16B per 1 KiB row
        const int col = (j & 63) << 4;       // byte offset within row
        const int goff = (tile * TT + row) * (E_ * 4) + col;   // byte off from chunk base
        // low 32 bits of a generic LDS pointer == LDS byte address
        const uint32_t lk = (uint32_t)(uintptr_t)(&sk[buf][row][0]) + (uint32_t)col;
        const uint32_t lv = (uint32_t)(uintptr_t)(&sv[buf][row][0]) + (uint32_t)col;
        asm volatile("global_load_async_to_lds_b128 %0, %1, %2"
                     :: "v"(lk), "v"(goff), "s"(kba) : "memory");
        asm volatile("global_load_async_to_lds_b128 %0, %1, %2"
                     :: "v"(lv), "v"(goff), "s"(vba) : "memory");
    }
}

// ---------------------------------------------------------------------------
// Pass 1: per-chunk summary scan.  State starts (0,0,-inf); the final
// (aa,bb,eps) represents C_A = aa*e^eps, C_B = bb*e^eps for the chunk.
// ---------------------------------------------------------------------------
__global__ __launch_bounds__(EW) void wkv_chunk_summary(
    const float* __restrict__ k, const float* __restrict__ v,
    const float* __restrict__ td_g,
    float* __restrict__ wsA, float* __restrict__ wsB, float* __restrict__ wsM)
{
    __shared__ float sk[2][TT][EW];
    __shared__ float sv[2][TT][EW];
    const int tid = threadIdx.x;
    int bx = blockIdx.x;
    const int ebi = bx % EB;  bx /= EB;
    const int b   = bx % B_;
    const int c   = bx / B_;
    const int eb  = ebi * EW;
    const size_t base = ((size_t)b * S_ + (size_t)c * LC) * E_ + eb;
    const float* kb = k + base;
    const float* vb = v + base;
    const float td = td_g[eb + tid];

    issue_tile(kb, vb, sk, sv, 0, 0, tid);

    float aa = 0.f, bb = 0.f, eps = -INFINITY;
#pragma unroll 1
    for (int t = 0; t < TILES; ++t) {
        const int buf = t & 1;
        if (t + 1 < TILES) {
            issue_tile(kb, vb, sk, sv, buf ^ 1, t + 1, tid);
            asm volatile("s_wait_asynccnt 0x8" ::: "memory");  // tile t landed
        } else {
            asm volatile("s_wait_asynccnt 0x0" ::: "memory");
        }
        __syncthreads();
#pragma unroll
        for (int r = 0; r < TT; ++r) {
            const float kt = sk[buf][r][tid];
            const float vt = sv[buf][r][tid];
            const float en = fmaxf(td + eps, kt);
            const float w  = __expf(td + eps - en);
            const float ek = __expf(kt - en);
            aa = aa * w + vt * ek;
            bb = bb * w + ek;
            eps = en;
        }
        __syncthreads();
    }
    const size_t idx = ((size_t)c * B_ + b) * E_ + eb + tid;
    wsA[idx] = aa; wsB[idx] = bb; wsM[idx] = eps;
}

// ---------------------------------------------------------------------------
// Pass 2: sequential combine of the NC summaries per channel -> exclusive
// prefix = chunk-start states, written in place over the summaries.
// Combine: A_start(c+1) = e^{L*td} * A_start(c) + C_A(c), log-stabilized.
// ---------------------------------------------------------------------------
__global__ __launch_bounds__(256) void wkv_combine(
    const float* __restrict__ td_g,
    float* __restrict__ wsA, float* __restrict__ wsB, float* __restrict__ wsM)
{
    const int gid = blockIdx.x * 256 + threadIdx.x;   // 0 .. B_*E_-1
    const int b = gid / E_;
    const int e = gid - b * E_;
    const float ltd = (float)LC * td_g[e];
    float aa = 0.f, bb = 0.f, eps = 0.f;              // matches reference init
#pragma unroll 1
    for (int c = 0; c < NC; ++c) {
        const size_t idx = ((size_t)c * B_ + b) * E_ + e;
        const float sA = wsA[idx], sB = wsB[idx], sM = wsM[idx];
        wsA[idx] = aa; wsB[idx] = bb; wsM[idx] = eps; // start state for chunk c
        const float m = fmaxf(eps + ltd, sM);
        const float w = __expf(eps + ltd - m);
        const float u = __expf(sM - m);
        aa = aa * w + sA * u;
        bb = bb * w + sB * u;
        eps = m;
    }
}

// ---------------------------------------------------------------------------
// Pass 3: replay the exact reference recurrence within each chunk from the
// chunk-start state, emitting y with non-temporal stores.
// ---------------------------------------------------------------------------
__global__ __launch_bounds__(EW) void wkv_output(
    const float* __restrict__ k, const float* __restrict__ v,
    const float* __restrict__ td_g, const float* __restrict__ tf_g,
    const float* __restrict__ wsA, const float* __restrict__ wsB,
    const float* __restrict__ wsM, float* __restrict__ y)
{
    __shared__ float sk[2][TT][EW];
    __shared__ float sv[2][TT][EW];
    const int tid = threadIdx.x;
    int bx = blockIdx.x;
    const int ebi = bx % EB;  bx /= EB;
    const int b   = bx % B_;
    const int c   = bx / B_;
    const int eb  = ebi * EW;
    const size_t base = ((size_t)b * S_ + (size_t)c * LC) * E_ + eb;
    const float* kb = k + base;
    const float* vb = v + base;
    float*       yb = y + base;
    const float td = td_g[eb + tid];
    const float tf = tf_g[eb + tid];
    const size_t sidx = ((size_t)c * B_ + b) * E_ + eb + tid;
    float aa = wsA[sidx], bb = wsB[sidx], eps = wsM[sidx];

    issue_tile(kb, vb, sk, sv, 0, 0, tid);

#pragma unroll 1
    for (int t = 0; t < TILES; ++t) {
        const int buf = t & 1;
        if (t + 1 < TILES) {
            issue_tile(kb, vb, sk, sv, buf ^ 1, t + 1, tid);
            asm volatile("s_wait_asynccnt 0x8" ::: "memory");
        } else {
            asm volatile("s_wait_asynccnt 0x0" ::: "memory");
        }
        __syncthreads();
        float* yrow = yb + (size_t)t * TT * E_ + tid;
#pragma unroll
        for (int r = 0; r < TT; ++r) {
            const float kt = sk[buf][r][tid];
            const float vt = sv[buf][r][tid];
            const float etfk = __expf(tf + kt - eps);
            const float yv   = (aa + vt * etfk) / (bb + etfk);
            __builtin_nontemporal_store(yv, yrow);
            yrow += E_;
            const float en = fmaxf(td + eps, kt);
            const float w  = __expf(td + eps - en);
            const float ek = __expf(kt - en);
            aa = aa * w + vt * ek;
            bb = bb * w + ek;
            eps = en;
        }
        __syncthreads();
    }
}

// ---------------------------------------------------------------------------
extern "C" void kernel_launch(void* const* d_in, const int* in_sizes, int n_in,
                              void* d_out, int out_size, void* d_ws, size_t ws_size,
                              hipStream_t stream)
{
    // setup_inputs order: batch_size, seq_len, embedding_dim, time_decay,
    //                     time_first, k, v
    const float* td = (const float*)d_in[3];
    const float* tf = (const float*)d_in[4];
    const float* k  = (const float*)d_in[5];
    const float* v  = (const float*)d_in[6];
    float* y = (float*)d_out;

    const size_t per = (size_t)NC * B_ * E_;          // 262144 floats per array
    float* wsA = (float*)d_ws;                        // needs 3 MiB of d_ws
    float* wsB = wsA + per;
    float* wsM = wsB + per;

    const dim3 gridScan(NC * B_ * EB);                // 1024 blocks of 256
    wkv_chunk_summary<<<gridScan, EW, 0, stream>>>(k, v, td, wsA, wsB, wsM);
    wkv_combine<<<dim3((B_ * E_) / 256), 256, 0, stream>>>(td, wsA, wsB, wsM);
    wkv_output<<<gridScan, EW, 0, stream>>>(k, v, td, tf, wsA, wsB, wsM, y);
}